// Linear_18476949307491
// MI455X (gfx1250) — compile-verified
//
#include <hip/hip_runtime.h>
#include <hip/hip_bf16.h>
#include <cstddef>

typedef __attribute__((ext_vector_type(2))) float v2f;
typedef __attribute__((ext_vector_type(4))) float v4f;
typedef __attribute__((ext_vector_type(8))) float v8f;

#define B_SZ  256
#define F_IN  1024
#define F_OUT 1024
#define OB 8   // output rows per block (one per wave)
#define BB 8   // batch rows per block (x tile in LDS = 32 KB)

__device__ __forceinline__ float softplus_f(float v) {
    // numerically stable softplus matching jax.nn.softplus
    return fmaxf(v, 0.0f) + log1pf(__expf(-fabsf(v)));
}

// ---------------------------------------------------------------------------
// Kernel A: sigma = softplus(weight_rho), 1M elements, vectorized.
// ---------------------------------------------------------------------------
__global__ __launch_bounds__(256) void sigma_kernel(const float* __restrict__ rho,
                                                    float* __restrict__ sigma, int n4) {
    int t = blockIdx.x * blockDim.x + threadIdx.x;
    if (t < n4) {
        v4f r = ((const v4f*)rho)[t];
        v4f s;
        s.x = softplus_f(r.x);
        s.y = softplus_f(r.y);
        s.z = softplus_f(r.z);
        s.w = softplus_f(r.w);
        ((v4f*)sigma)[t] = s;
    }
}

// ---------------------------------------------------------------------------
// Kernel B: y0[b,o] = sum_i x[b,i] * mu[o,i]  (fp32 GEMM via V_WMMA_F32_16X16X4_F32)
// One wave per 16x16 output tile; 256 WMMA steps over K=1024.
// A (16x4 f32): VGPR0 = K{0,2}, VGPR1 = K{1,3}, lanes 0-15 / 16-31 split.
// B (4x16 f32): mirrored row striping; C (16x16 f32): VGPR r -> M=r / r+8.
// ---------------------------------------------------------------------------
__global__ __launch_bounds__(32) void gemm_mu_wmma(const float* __restrict__ x,
                                                   const float* __restrict__ mu,
                                                   float* __restrict__ y0) {
    const int tile = blockIdx.x;
    const int tm = tile / (F_OUT / 16);   // batch tile
    const int tn = tile % (F_OUT / 16);   // out-feature tile
    const int lane = threadIdx.x;
    const int half = lane >> 4;           // 0 or 1
    const int l16  = lane & 15;

    const float* xrow  = x  + (size_t)(tm * 16 + l16) * F_IN;  // A row m = l16
    const float* murow = mu + (size_t)(tn * 16 + l16) * F_IN;  // B col n = l16

    v8f c = {};
    for (int k = 0; k < F_IN; k += 4) {
        v2f a, b;
        a.x = xrow[k + 2 * half + 0];
        a.y = xrow[k + 2 * half + 1];
        b.x = murow[k + 2 * half + 0];
        b.y = murow[k + 2 * half + 1];
        c = __builtin_amdgcn_wmma_f32_16x16x4_f32(
            /*neg_a=*/false, a, /*neg_b=*/false, b,
            /*c_mod=*/(short)0, c, /*reuse_a=*/false, /*reuse_b=*/false);
    }

    float* outp = y0 + (size_t)tn * 16 + l16;
#pragma unroll
    for (int r = 0; r < 8; ++r) {
        int m = tm * 16 + r + 8 * half;
        outp[(size_t)m * F_OUT] = c[r];
    }
}

// ---------------------------------------------------------------------------
// Kernel C: bandwidth kernel. Streams eps_w (1 GB) exactly once from HBM with
// NON-TEMPORAL loads so the one-shot stream does not evict sigma/x/y0 from L2.
// Block: 256 threads = 8 waves. Wave w owns output row o = blk.x*8 + w,
// iterates over 8 batch rows. sigma row lives in 32 VGPRs/lane; x tile in LDS.
// Inner loop: 1x global_load_b128 (eps, NT) + 1x ds_load_b128 (x) + 4 mul + 4 fma.
// ---------------------------------------------------------------------------
__global__ __launch_bounds__(256) void eps_kernel(const float* __restrict__ x,
                                                  const float* __restrict__ sigma,
                                                  const float* __restrict__ y0,
                                                  const float* __restrict__ bias_mu,
                                                  const float* __restrict__ bias_rho,
                                                  const float* __restrict__ eps_w,
                                                  const float* __restrict__ eps_b,
                                                  float* __restrict__ out) {
    __shared__ float xs[BB * F_IN];  // 32 KB
    const int lane = threadIdx.x & 31;
    const int wave = threadIdx.x >> 5;
    const int b0 = blockIdx.y * BB;
    const int o  = blockIdx.x * OB + wave;

    // cooperative x-tile load: BB*F_IN/4 = 2048 float4 over 256 threads
    const v4f* xg = (const v4f*)(x + (size_t)b0 * F_IN);
    for (int t = threadIdx.x; t < BB * F_IN / 4; t += 256) {
        ((v4f*)xs)[t] = xg[t];
    }
    __syncthreads();

    // sigma row -> registers (reused across all BB batch rows; stays L2-hot)
    const float* srow = sigma + (size_t)o * F_IN;
    v4f sg[8];
#pragma unroll
    for (int it = 0; it < 8; ++it)
        sg[it] = *(const v4f*)(srow + it * 128 + lane * 4);

    const float bmu = bias_mu[o];
    const float sb  = softplus_f(bias_rho[o]);

    for (int bb = 0; bb < BB; ++bb) {
        const int b = b0 + bb;
        const float* erow = eps_w + ((size_t)b * F_OUT + o) * F_IN;
        float acc = 0.0f;
#pragma unroll
        for (int it = 0; it < 8; ++it) {
            const int i = it * 128 + lane * 4;
            v4f e  = __builtin_nontemporal_load((const v4f*)(erow + i)); // HBM stream, TH=NT
            v4f xv = *(const v4f*)(xs + bb * F_IN + i);                  // LDS
            acc = fmaf(e.x, xv.x * sg[it].x, acc);
            acc = fmaf(e.y, xv.y * sg[it].y, acc);
            acc = fmaf(e.z, xv.z * sg[it].z, acc);
            acc = fmaf(e.w, xv.w * sg[it].w, acc);
        }
        // wave32 tree reduction
#pragma unroll
        for (int m = 16; m > 0; m >>= 1)
            acc += __shfl_xor(acc, m, 32);
        if (lane == 0) {
            const size_t idx = (size_t)b * F_OUT + o;
            float r = acc + y0[idx] + bmu + sb * eps_b[idx];
            __builtin_nontemporal_store(r, &out[idx]);   // written once, never re-read
        }
    }
}

// ---------------------------------------------------------------------------
extern "C" void kernel_launch(void* const* d_in, const int* in_sizes, int n_in,
                              void* d_out, int out_size, void* d_ws, size_t ws_size,
                              hipStream_t stream) {
    const float* x     = (const float*)d_in[0];
    const float* wmu   = (const float*)d_in[1];
    const float* wrho  = (const float*)d_in[2];
    const float* bmu   = (const float*)d_in[3];
    const float* brho  = (const float*)d_in[4];
    const float* eps_w = (const float*)d_in[5];
    const float* eps_b = (const float*)d_in[6];
    float* out = (float*)d_out;

    // workspace layout: sigma (4 MB) | y0 (1 MB)
    float* sigma = (float*)d_ws;
    float* y0    = sigma + (size_t)F_OUT * F_IN;

    const int n4 = F_OUT * F_IN / 4;
    sigma_kernel<<<(n4 + 255) / 256, 256, 0, stream>>>(wrho, sigma, n4);

    gemm_mu_wmma<<<(B_SZ / 16) * (F_OUT / 16), 32, 0, stream>>>(x, wmu, y0);

    dim3 grid(F_OUT / OB, B_SZ / BB);
    eps_kernel<<<grid, 256, 0, stream>>>(x, sigma, y0, bmu, brho, eps_w, eps_b, out);
}